// GCNRoutePredictor_74131135529943
// MI455X (gfx1250) — compile-verified
//
#include <hip/hip_runtime.h>
#include <hip/hip_bf16.h>
#include <cstdint>

typedef __attribute__((ext_vector_type(2))) float v2f;
typedef __attribute__((ext_vector_type(8))) float v8f;

#define FDIM 128
#define ROWS_PER_BLOCK 64   // 4 row tiles of 16 per wave

// ---------------------------------------------------------------------------
// gfx1250 async copy: global -> LDS, 16 bytes per lane, tracked by ASYNCcnt.
// LDS address = low 32 bits of the generic shared pointer (LDS aperture).
// ---------------------------------------------------------------------------
__device__ __forceinline__ void async_copy_b128(unsigned lds_addr, const void* gptr) {
    asm volatile("global_load_async_to_lds_b128 %0, %1, off"
                 :: "v"(lds_addr), "v"((unsigned long long)(uintptr_t)gptr)
                 : "memory");
}
__device__ __forceinline__ void wait_async_zero() {
    asm volatile("s_wait_asynccnt 0x0" ::: "memory");
}

// ---------------------------------------------------------------------------
// Degree computation: deg = 1 + in-degree; dinv = rsqrt(deg)
// ---------------------------------------------------------------------------
__global__ __launch_bounds__(256) void k_deg_init(float* __restrict__ deg, int n) {
    int i = blockIdx.x * blockDim.x + threadIdx.x;
    if (i < n) deg[i] = 1.0f;
}

__global__ __launch_bounds__(256) void k_deg_count(const int* __restrict__ dst,
                                                   float* __restrict__ deg, int ne) {
    int e = blockIdx.x * blockDim.x + threadIdx.x;
    if (e < ne) atomicAdd(&deg[dst[e]], 1.0f);
}

__global__ __launch_bounds__(256) void k_deg_rsqrt(const float* __restrict__ deg,
                                                   float* __restrict__ dinv, int n) {
    int i = blockIdx.x * blockDim.x + threadIdx.x;
    if (i < n) dinv[i] = rsqrtf(deg[i]);
}

// ---------------------------------------------------------------------------
// Fp32 WMMA GEMM: H[n,128] = X[n,128] @ W[128,128]
//  - W (64 KB) staged into LDS once per block via global_load_async_to_lds_b128
//  - block = 256 threads = 8 waves; wave w owns column tile n0 = w*16 and
//    FOUR row tiles (64 rows per block) -> 4x B-fragment reuse, 128 WMMAs/wave
//
// Fragment layouts (CDNA5 ISA 7.12.2):
//   A 16x4 f32 : lanes 0-15 -> M=lane, v0=K0,v1=K1 ; lanes 16-31 -> v0=K2,v1=K3
//   B 4x16 f32 : VGPR r, lanes 0-15 -> K=r row (N=lane) ; lanes 16-31 -> K=r+2
//   C/D 16x16  : VGPR r, lanes 0-15 -> M=r ; lanes 16-31 -> M=r+8 ; N=lane&15
// ---------------------------------------------------------------------------
__global__ __launch_bounds__(256) void k_gemm128_wmma(const float* __restrict__ X,
                                                      const float* __restrict__ W,
                                                      float* __restrict__ H, int n_rows) {
    __shared__ float lw[FDIM * FDIM];              // 64 KB weight tile

    const int lane  = threadIdx.x & 31;
    const int wv    = threadIdx.x >> 5;            // 0..7 -> column tile
    const int n0    = wv * 16;
    const int half  = lane >> 4;                   // 0: K pair {0,1}, 1: {2,3}
    const int l15   = lane & 15;
    const int mbase = blockIdx.x * ROWS_PER_BLOCK;

    // ---- stage W into LDS (4096 float4 ops across 256 threads) ----
    {
        const unsigned lbase = (unsigned)(uintptr_t)&lw[0];
#pragma unroll
        for (int it = 0; it < 16; ++it) {
            const int j = it * 256 + threadIdx.x;  // float4 index
            async_copy_b128(lbase + (unsigned)j * 16u, W + (size_t)j * 4);
        }
        wait_async_zero();
        __syncthreads();
    }

    // ---- A row pointers for the 4 row tiles (clamped for tail block) ----
    const float* Arow[4];
#pragma unroll
    for (int mt = 0; mt < 4; ++mt) {
        int r = mbase + mt * 16 + l15;
        if (r >= n_rows) r = n_rows - 1;
        Arow[mt] = X + (size_t)r * FDIM;
    }

    v8f c[4];
#pragma unroll
    for (int mt = 0; mt < 4; ++mt) c[mt] = (v8f){0.f,0.f,0.f,0.f,0.f,0.f,0.f,0.f};

#pragma unroll
    for (int k = 0; k < FDIM; k += 4) {
        const int kb = k + half * 2;
        v2f b;
        b.x = lw[kb * FDIM + n0 + l15];
        b.y = lw[(kb + 1) * FDIM + n0 + l15];
#pragma unroll
        for (int mt = 0; mt < 4; ++mt) {
            v2f a;
            a.x = Arow[mt][kb];
            a.y = Arow[mt][kb + 1];
            // (neg_a, A, neg_b, B, c_mod, C, reuse_a, reuse_b)
            c[mt] = __builtin_amdgcn_wmma_f32_16x16x4_f32(false, a, false, b,
                                                          (short)0, c[mt], false, false);
        }
    }

    // ---- store D tiles ----
    const int col = n0 + l15;
#pragma unroll
    for (int mt = 0; mt < 4; ++mt) {
#pragma unroll
        for (int r = 0; r < 8; ++r) {
            const int row = mbase + mt * 16 + r + half * 8;
            if (row < n_rows) H[(size_t)row * FDIM + col] = c[mt][r];
        }
    }
}

// ---------------------------------------------------------------------------
// agg init: agg[i,f] = h[i,f] * dinv[i]^2 + bias[f]   (self-loop term + bias)
// ---------------------------------------------------------------------------
__global__ __launch_bounds__(256) void k_init_agg(const float* __restrict__ h,
                                                  const float* __restrict__ dinv,
                                                  const float* __restrict__ bias,
                                                  float* __restrict__ agg, int n) {
    int i = blockIdx.x * blockDim.x + threadIdx.x;
    int total = n * FDIM;
    if (i < total) {
        int node = i >> 7;
        int f    = i & (FDIM - 1);
        float d  = dinv[node];
        agg[i] = h[i] * d * d + bias[f];
    }
}

// ---------------------------------------------------------------------------
// Edge scatter: one wave per edge; lane l handles features [4l..4l+3].
// agg[dst] += h[src] * (dinv[src]*dinv[dst])
// Feature buffers (25.6 MB) fit in 192 MB L2 -> atomics resolve in L2.
// ---------------------------------------------------------------------------
__global__ __launch_bounds__(256) void k_edge_scatter(const float* __restrict__ h,
                                                      const int* __restrict__ src,
                                                      const int* __restrict__ dst,
                                                      const float* __restrict__ dinv,
                                                      float* __restrict__ agg, int ne) {
    int t = blockIdx.x * blockDim.x + threadIdx.x;
    int e = t >> 5;
    if (e >= ne) return;
    int lane = threadIdx.x & 31;
    int s = src[e];
    int d = dst[e];
    float norm = dinv[s] * dinv[d];
    const float4 v = ((const float4*)(h + (size_t)s * FDIM))[lane];
    float* ad = agg + (size_t)d * FDIM + lane * 4;
    atomicAdd(ad + 0, v.x * norm);
    atomicAdd(ad + 1, v.y * norm);
    atomicAdd(ad + 2, v.z * norm);
    atomicAdd(ad + 3, v.w * norm);
}

// ---------------------------------------------------------------------------
// Fused LayerNorm + ReLU, in place. One wave (32 lanes) per node row of 128.
// ---------------------------------------------------------------------------
__global__ __launch_bounds__(256) void k_ln_relu(float* __restrict__ h,
                                                 const float* __restrict__ gamma,
                                                 const float* __restrict__ beta, int n) {
    int node = (blockIdx.x * blockDim.x + threadIdx.x) >> 5;
    if (node >= n) return;
    int lane = threadIdx.x & 31;

    float4* rowp = (float4*)(h + (size_t)node * FDIM);
    float4 v = rowp[lane];

    float s = v.x + v.y + v.z + v.w;
#pragma unroll
    for (int off = 16; off >= 1; off >>= 1) s += __shfl_xor(s, off, 32);
    float mu = s * (1.0f / FDIM);

    float dx = v.x - mu, dy = v.y - mu, dz = v.z - mu, dw = v.w - mu;
    float q = dx * dx + dy * dy + dz * dz + dw * dw;
#pragma unroll
    for (int off = 16; off >= 1; off >>= 1) q += __shfl_xor(q, off, 32);
    float inv = rsqrtf(q * (1.0f / FDIM) + 1e-5f);

    float4 g = ((const float4*)gamma)[lane];
    float4 b = ((const float4*)beta)[lane];
    float4 o;
    o.x = fmaxf(dx * inv * g.x + b.x, 0.0f);
    o.y = fmaxf(dy * inv * g.y + b.y, 0.0f);
    o.z = fmaxf(dz * inv * g.z + b.z, 0.0f);
    o.w = fmaxf(dw * inv * g.w + b.w, 0.0f);
    rowp[lane] = o;
}

// ---------------------------------------------------------------------------
// Layer-3 GEMV + self-loop: z[i] = dot(h[i,:], W3); out[i] = z[i]*dinv^2 + b3
// One wave per node.
// ---------------------------------------------------------------------------
__global__ __launch_bounds__(256) void k_gemv_out(const float* __restrict__ h,
                                                  const float* __restrict__ W3,
                                                  const float* __restrict__ b3,
                                                  const float* __restrict__ dinv,
                                                  float* __restrict__ z,
                                                  float* __restrict__ out, int n) {
    int node = (blockIdx.x * blockDim.x + threadIdx.x) >> 5;
    if (node >= n) return;
    int lane = threadIdx.x & 31;

    float4 v = ((const float4*)(h + (size_t)node * FDIM))[lane];
    float4 w = ((const float4*)W3)[lane];
    float s = v.x * w.x + v.y * w.y + v.z * w.z + v.w * w.w;
#pragma unroll
    for (int off = 16; off >= 1; off >>= 1) s += __shfl_xor(s, off, 32);

    if (lane == 0) {
        float d = dinv[node];
        z[node]   = s;
        out[node] = s * d * d + b3[0];
    }
}

__global__ __launch_bounds__(256) void k_edge_scatter_scalar(const float* __restrict__ z,
                                                             const int* __restrict__ src,
                                                             const int* __restrict__ dst,
                                                             const float* __restrict__ dinv,
                                                             float* __restrict__ out, int ne) {
    int e = blockIdx.x * blockDim.x + threadIdx.x;
    if (e < ne) {
        int s = src[e], d = dst[e];
        atomicAdd(&out[d], z[s] * dinv[s] * dinv[d]);
    }
}

// ---------------------------------------------------------------------------
// Host-side orchestration (all on `stream`, graph-capture safe).
// ---------------------------------------------------------------------------
extern "C" void kernel_launch(void* const* d_in, const int* in_sizes, int n_in,
                              void* d_out, int out_size, void* d_ws, size_t ws_size,
                              hipStream_t stream) {
    const float* x     = (const float*)d_in[0];
    const int*   eidx  = (const int*)d_in[1];
    const float* W1    = (const float*)d_in[2];
    const float* b1    = (const float*)d_in[3];
    const float* W2    = (const float*)d_in[4];
    const float* b2    = (const float*)d_in[5];
    const float* W3    = (const float*)d_in[6];
    const float* b3    = (const float*)d_in[7];
    const float* gamma = (const float*)d_in[8];
    const float* beta  = (const float*)d_in[9];
    float*       out   = (float*)d_out;

    const int n  = in_sizes[0] / FDIM;   // 50000 nodes
    const int ne = in_sizes[1] / 2;      // 600000 edges
    const int* src = eidx;
    const int* dst = eidx + ne;

    // Workspace layout (floats)
    float* ws   = (float*)d_ws;
    const size_t feat = (size_t)n * FDIM;
    float* t0   = ws;                    // [n,128] GEMM result
    float* t1   = t0 + feat;             // [n,128] aggregation / LN buffer
    float* deg  = t1 + feat;             // [n]
    float* dinv = deg + n;               // [n]
    float* z    = dinv + n;              // [n]

    const int TB = 256;
    dim3 blk(TB);
    dim3 g_nodes((n + TB - 1) / TB);
    dim3 g_edges((ne + TB - 1) / TB);
    dim3 g_feat((int)((feat + TB - 1) / TB));
    dim3 g_gemm((n + ROWS_PER_BLOCK - 1) / ROWS_PER_BLOCK);   // 64 rows / block
    dim3 g_wave_node((n * 32 + TB - 1) / TB);                 // 1 wave per node
    dim3 g_wave_edge(((size_t)ne * 32 + TB - 1) / TB);        // 1 wave per edge

    // Degrees
    k_deg_init<<<g_nodes, blk, 0, stream>>>(deg, n);
    k_deg_count<<<g_edges, blk, 0, stream>>>(dst, deg, ne);
    k_deg_rsqrt<<<g_nodes, blk, 0, stream>>>(deg, dinv, n);

    // Layer 1
    k_gemm128_wmma<<<g_gemm, blk, 0, stream>>>(x, W1, t0, n);
    k_init_agg<<<g_feat, blk, 0, stream>>>(t0, dinv, b1, t1, n);
    k_edge_scatter<<<g_wave_edge, blk, 0, stream>>>(t0, src, dst, dinv, t1, ne);
    k_ln_relu<<<g_wave_node, blk, 0, stream>>>(t1, gamma, beta, n);

    // Layer 2
    k_gemm128_wmma<<<g_gemm, blk, 0, stream>>>(t1, W2, t0, n);
    k_init_agg<<<g_feat, blk, 0, stream>>>(t0, dinv, b2, t1, n);
    k_edge_scatter<<<g_wave_edge, blk, 0, stream>>>(t0, src, dst, dinv, t1, ne);
    k_ln_relu<<<g_wave_node, blk, 0, stream>>>(t1, gamma, beta, n);

    // Layer 3 (GEMV + aggregation into d_out)
    k_gemv_out<<<g_wave_node, blk, 0, stream>>>(t1, W3, b3, dinv, z, out, n);
    k_edge_scatter_scalar<<<g_edges, blk, 0, stream>>>(z, src, dst, dinv, out, ne);
}